// LabelAwareAttention_26508538150968
// MI455X (gfx1250) — compile-verified
//
#include <hip/hip_runtime.h>

// LabelAwareAttention fused kernel for MI455X (gfx1250, wave32).
// B=4096, C=8, S=200, K=64. Memory-bound (~430MB @ 23.3TB/s => ~18us floor),
// so: one fused kernel, scores/weights live in VGPRs/LDS only.
// Matrix ops use V_WMMA_F32_16X16X4_F32 (fp32 WMMA, 16 k-steps for K=64).

typedef float v2f __attribute__((ext_vector_type(2)));
typedef float v4f __attribute__((ext_vector_type(4)));
typedef float v8f __attribute__((ext_vector_type(8)));

#define C_DIM 8
#define K_DIM 64
#define WAVES 4

__global__ __launch_bounds__(128) void laa_kernel(
    const float* __restrict__ caps,          // [B, 8, 64]
    const unsigned char* __restrict__ mask,  // [B, 8] (bool)
    const float* __restrict__ X,             // [B, S, 64]
    float* __restrict__ out,                 // [B, S, 64]
    int S)
{
    __shared__ float sCaps[16 * K_DIM];        // capsules, zero-padded 16x64
    __shared__ float sX[WAVES][16 * K_DIM];    // per-wave X tile (16 s-rows)
    __shared__ float sW[WAVES][16 * 8];        // per-wave softmax weights (s x c)

    const int b    = blockIdx.x;
    const int tid  = threadIdx.x;
    const int wave = tid >> 5;
    const int lane = tid & 31;
    const int l16  = lane & 15;
    const int hi2  = (lane >> 4) << 1;   // 0 for lanes 0-15, 2 for lanes 16-31

    // ---- stage capsules into LDS (rows 8..15 zero-padded) ----
    {
        const float* gc = caps + (size_t)b * (C_DIM * K_DIM);
        const int base = tid * 8;                    // 128 thr * 8 = 1024 floats
        if (base < C_DIM * K_DIM) {
            *(v4f*)&sCaps[base]     = *(const v4f*)&gc[base];
            *(v4f*)&sCaps[base + 4] = *(const v4f*)&gc[base + 4];
        } else {
            v4f z = {};
            *(v4f*)&sCaps[base]     = z;
            *(v4f*)&sCaps[base + 4] = z;
        }
    }
    // per-batch capsule mask: 8 bytes, one per capsule
    const unsigned long long mword =
        *(const unsigned long long*)(mask + (size_t)b * C_DIM);
    __syncthreads();

    const int ntiles = (S + 15) >> 4;
    float* myX = &sX[wave][0];
    float* myW = &sW[wave][0];
    const float* gX = X + (size_t)b * S * K_DIM;
    float*       gO = out + (size_t)b * S * K_DIM;

    for (int t = wave; t < ntiles; t += WAVES) {
        const int s0 = t << 4;

        // ---- stage X tile (16x64 f32) into wave-private LDS, pad past S ----
        #pragma unroll
        for (int q = 0; q < 8; ++q) {
            const int flat = q * 128 + lane * 4;
            const int row  = flat >> 6;
            const int col  = flat & 63;
            v4f v = {};
            if (s0 + row < S) v = *(const v4f*)&gX[(size_t)(s0 + row) * K_DIM + col];
            *(v4f*)&myX[row * K_DIM + col] = v;
        }
        // prefetch next tile of X into cache (global_prefetch_b8)
        if (t + WAVES < ntiles)
            __builtin_prefetch(&gX[(size_t)((t + WAVES) << 4) * K_DIM + lane * 8], 0, 0);

        // wave-private LDS: DS ops are in-order within a wave, just fence issue
        asm volatile("s_wait_dscnt 0" ::: "memory");

        // ---- pass 1: scores(16c x 16s) = caps(16x64) x X^T(64x16) ----
        // A 16x4 f32 layout: lane%16 = row M; VGPR0/1 hold K = 4i+2*hi +{0,1}
        // B 4x16 f32 layout: lane%16 = col N (=s); VGPR0/1 = rows K (same split)
        v8f acc = {};
        #pragma unroll
        for (int i = 0; i < 16; ++i) {
            const int k = 4 * i + hi2;
            v2f a  = *(const v2f*)&sCaps[l16 * K_DIM + k];
            v2f bm = *(const v2f*)&myX[l16 * K_DIM + k];
            acc = __builtin_amdgcn_wmma_f32_16x16x4_f32(
                false, a, false, bm, (short)0, acc, false, false);
        }

        // ---- mask + softmax over capsules (lanes 0-15: VGPR j == c=j,
        //      column s = s0 + lane; lanes 16-31 hold zero-pad rows) ----
        float sc[8];
        #pragma unroll
        for (int j = 0; j < 8; ++j) {
            const bool drop = ((mword >> (8 * j)) & 0xffULL) != 0ULL; // mask True -> drop
            sc[j] = drop ? -3.4028235e38f : acc[j];
        }
        float m = sc[0];
        #pragma unroll
        for (int j = 1; j < 8; ++j) m = fmaxf(m, sc[j]);
        float sum = 0.f;
        #pragma unroll
        for (int j = 0; j < 8; ++j) { sc[j] = __expf(sc[j] - m); sum += sc[j]; }
        const float inv = 1.f / sum;

        if (lane < 16) {   // only lanes 0-15 hold real capsule rows
            #pragma unroll
            for (int j = 0; j < 4; ++j) {
                v2f w2 = { sc[2 * j] * inv, sc[2 * j + 1] * inv };
                *(v2f*)&myW[l16 * 8 + 2 * j] = w2;     // [s][c] row-major
            }
        }
        asm volatile("s_wait_dscnt 0" ::: "memory");

        // ---- pass 2: out(16s x 64k) = W(16x8) x caps(8x64) ----
        #pragma unroll
        for (int kt = 0; kt < 4; ++kt) {
            v8f o = {};
            #pragma unroll
            for (int cs = 0; cs < 2; ++cs) {
                const int c0 = 4 * cs + hi2;
                v2f a2 = *(const v2f*)&myW[l16 * 8 + c0];
                v2f b2;
                b2.x = sCaps[(c0    ) * K_DIM + kt * 16 + l16];
                b2.y = sCaps[(c0 + 1) * K_DIM + kt * 16 + l16];
                o = __builtin_amdgcn_wmma_f32_16x16x4_f32(
                    false, a2, false, b2, (short)0, o, false, false);
            }
            // D layout: VGPR j -> row s0+j (lanes 0-15) / s0+j+8 (lanes 16-31)
            const int hi8 = (lane >> 4) << 3;
            #pragma unroll
            for (int j = 0; j < 8; ++j) {
                const int srow = s0 + j + hi8;
                if (srow < S)
                    gO[(size_t)srow * K_DIM + kt * 16 + l16] = o[j];
            }
        }
    }
}

extern "C" void kernel_launch(void* const* d_in, const int* in_sizes, int n_in,
                              void* d_out, int out_size, void* d_ws, size_t ws_size,
                              hipStream_t stream) {
    const float*         caps = (const float*)d_in[0];
    const unsigned char* mask = (const unsigned char*)d_in[1];
    const float*         X    = (const float*)d_in[2];
    float*               out  = (float*)d_out;

    const int B = in_sizes[1] / C_DIM;                 // mask is [B, C]
    const int S = in_sizes[2] / (B * K_DIM);           // X is [B, S, K]

    laa_kernel<<<dim3(B), dim3(128), 0, stream>>>(caps, mask, X, out, S);
}